// SVRasterGPU_26422638805065
// MI455X (gfx1250) — compile-verified
//
#include <hip/hip_runtime.h>
#include <stdint.h>

// SVRaster: per-ray DDA traversal of a 16^3 voxel grid with front-to-back
// alpha compositing. CDNA5 paths:
//   - TDM tensor_load_to_lds DMA of the 48KB color table (TENSORcnt)
//   - per-lane global_load_async_to_lds_b128 of the 16KB density table (ASYNCcnt)
//   - split wait counters (s_wait_asynccnt / s_wait_tensorcnt)

#define BASE_RES   16
#define NVOX       (BASE_RES * BASE_RES * BASE_RES)   // 4096
#define MAX_HITS   100
#define RAY_EPS    1e-8f
#define SCENE_MIN  (-1.0f)
#define SCENE_MAX  (1.0f)
#define CELL_H     ((SCENE_MAX - SCENE_MIN) / (float)BASE_RES)  // 0.125

typedef uint32_t u32x4 __attribute__((ext_vector_type(4)));
typedef uint32_t u32x8 __attribute__((ext_vector_type(8)));

__device__ __forceinline__ float safe_inv(float d) {
    float s = (fabsf(d) < RAY_EPS) ? ((d >= 0.0f) ? RAY_EPS : -RAY_EPS) : d;
    return 1.0f / s;
}

__global__ __launch_bounds__(256) void svraster_dda_kernel(
    const float* __restrict__ ray_o,      // [R,3]
    const float* __restrict__ ray_d,      // [R,3]
    const float* __restrict__ vpos,       // [V,3]
    const float* __restrict__ vsize,      // [V]
    const float* __restrict__ vdens,      // [V]
    const float* __restrict__ vcol,       // [V,3]
    float* __restrict__ out_rgb,          // [R,3]
    float* __restrict__ out_depth,        // [R]
    int*   __restrict__ out_counts,       // [R]
    int*   __restrict__ out_idx,          // [R,MAX_HITS]
    int R)
{
    // 16KB densities + 48KB colors = 64KB LDS per workgroup (WGP has 320KB).
    __shared__ __align__(16) float sDens[NVOX];
    __shared__ __align__(16) float sCol[NVOX * 3];

    const int tid = threadIdx.x;

    // ---- Stage density table: per-lane async DMA (ASYNCcnt) ---------------
    {
        uint64_t gbase_d = (uint64_t)(uintptr_t)vdens;
        uint32_t lds_d   = (uint32_t)(uintptr_t)&sDens[0];   // low 32b == LDS offset
        #pragma unroll
        for (int i = tid; i < NVOX / 4; i += 256) {          // 1024 x b128
            uint32_t goff = (uint32_t)i * 16u;
            uint32_t loff = lds_d + (uint32_t)i * 16u;
            asm volatile("global_load_async_to_lds_b128 %0, %1, %2 offset:0"
                         :: "v"(loff), "v"(goff), "s"(gbase_d) : "memory");
        }
    }

    // ---- Stage color table: one TDM DMA issued by wave 0 (TENSORcnt) ------
    // Uniform branch (readfirstlane) so waves 1..7 truly skip issue: TDM
    // ignores EXEC, an exec-mask guard alone would issue 8 redundant DMAs.
    if (__builtin_amdgcn_readfirstlane(tid) == 0) {
        const uint64_t ga    = (uint64_t)(uintptr_t)vcol;
        const uint32_t lds_c = (uint32_t)(uintptr_t)&sCol[0];
        const uint32_t nelem = (uint32_t)(NVOX * 3);         // 12288 f32 = 48KB

        // D# group 0: count=1 | lds_addr | global_addr[56:0] | type=2
        const u32x4 g0 = {
            1u,                                               // count=1, user mode
            lds_c,                                            // lds_addr (bytes)
            (uint32_t)(ga & 0xffffffffu),                     // global_addr[31:0]
            (uint32_t)((ga >> 32) & 0x1ffffffu) | (2u << 30)  // addr[56:32] | type=2
        };
        // D# group 1: wg_mask=0, data_size=4B, 1-row 2D tensor:
        //   tensor_dim0 = tile_dim0 = 12288, tensor_dim1 = tile_dim1 = 1,
        //   tensor_dim0_stride = 12288. No padding, no iterate, no gather.
        const u32x8 g1 = {
            (2u << 16),                                       // data_size = 4 bytes
            (nelem & 0xffffu) << 16,                          // tensor_dim0[15:0]
            ((nelem >> 16) & 0xffffu) | (1u << 16),           // dim0[31:16] | tensor_dim1=1
            (nelem & 0xffffu) << 16,                          // tile_dim0 = 12288
            1u,                                               // tile_dim1=1, tile_dim2=0
            nelem,                                            // tensor_dim0_stride[31:0]
            0u,                                               // stride0 hi | stride1 lo
            0u                                                // stride1 hi
        };
        // 2-operand form: VADDR2/VADDR3 = NULL (tensor rank <= 2).
        asm volatile("tensor_load_to_lds %0, %1"
                     :: "s"(g0), "s"(g1) : "memory");
        __builtin_amdgcn_s_wait_tensorcnt(0);                 // drain TDM
    }

    asm volatile("s_wait_asynccnt 0x0" ::: "memory");         // drain async queue
    __syncthreads();                                          // publish LDS to all waves

    const int r = blockIdx.x * blockDim.x + tid;
    if (r >= R) return;

    __builtin_prefetch(ray_o + (size_t)r * 3, 0, 0);   // global_prefetch_b8
    __builtin_prefetch(ray_d + (size_t)r * 3, 0, 0);

    const float ox = ray_o[r * 3 + 0], oy = ray_o[r * 3 + 1], oz = ray_o[r * 3 + 2];
    const float dx = ray_d[r * 3 + 0], dy = ray_d[r * 3 + 1], dz = ray_d[r * 3 + 2];
    const float ivx = safe_inv(dx), ivy = safe_inv(dy), ivz = safe_inv(dz);

    // Ray vs scene AABB
    float tx0 = (SCENE_MIN - ox) * ivx, tx1 = (SCENE_MAX - ox) * ivx;
    float ty0 = (SCENE_MIN - oy) * ivy, ty1 = (SCENE_MAX - oy) * ivy;
    float tz0 = (SCENE_MIN - oz) * ivz, tz1 = (SCENE_MAX - oz) * ivz;
    float tEnter = fmaxf(fmaxf(fminf(tx0, tx1), fminf(ty0, ty1)), fminf(tz0, tz1));
    float tExit  = fminf(fminf(fmaxf(tx0, tx1), fmaxf(ty0, ty1)), fmaxf(tz0, tz1));
    tEnter = fmaxf(tEnter, 0.0f);

    float accR = 0.0f, accG = 0.0f, accB = 0.0f, accD = 0.0f, T = 1.0f;
    int   cnt = 0;
    int* __restrict__ myidx = out_idx + (size_t)r * MAX_HITS;

    if (tExit > tEnter) {
        // Entry cell (tiny nudge off the boundary)
        const float tN = tEnter + 1e-6f;
        int cx = (int)floorf((ox + dx * tN - SCENE_MIN) / CELL_H);
        int cy = (int)floorf((oy + dy * tN - SCENE_MIN) / CELL_H);
        int cz = (int)floorf((oz + dz * tN - SCENE_MIN) / CELL_H);
        cx = min(max(cx, 0), BASE_RES - 1);
        cy = min(max(cy, 0), BASE_RES - 1);
        cz = min(max(cz, 0), BASE_RES - 1);

        const int sx = (dx > 0.0f) ? 1 : -1;
        const int sy = (dy > 0.0f) ? 1 : -1;
        const int sz = (dz > 0.0f) ? 1 : -1;
        const float tDx = fabsf(CELL_H * ivx);
        const float tDy = fabsf(CELL_H * ivy);
        const float tDz = fabsf(CELL_H * ivz);
        float tMx = (SCENE_MIN + (cx + (sx > 0 ? 1 : 0)) * CELL_H - ox) * ivx;
        float tMy = (SCENE_MIN + (cy + (sy > 0 ? 1 : 0)) * CELL_H - oy) * ivy;
        float tMz = (SCENE_MIN + (cz + (sz > 0 ? 1 : 0)) * CELL_H - oz) * ivz;

        // DDA visits voxels in increasing t_near order -> no sort needed.
        for (int it = 0; it < 3 * BASE_RES + 4; ++it) {
            const int v = (cx * BASE_RES + cy) * BASE_RES + cz;

            // Exact slab test against the voxel's stored position/size.
            const float pcx = vpos[v * 3 + 0];
            const float pcy = vpos[v * 3 + 1];
            const float pcz = vpos[v * 3 + 2];
            const float hh  = vsize[v] * 0.5f;
            float a0 = (pcx - hh - ox) * ivx, a1 = (pcx + hh - ox) * ivx;
            float tn = fminf(a0, a1),        tf = fmaxf(a0, a1);
            float b0 = (pcy - hh - oy) * ivy, b1 = (pcy + hh - oy) * ivy;
            tn = fmaxf(tn, fminf(b0, b1));   tf = fminf(tf, fmaxf(b0, b1));
            float c0 = (pcz - hh - oz) * ivz, c1 = (pcz + hh - oz) * ivz;
            tn = fmaxf(tn, fminf(c0, c1));   tf = fminf(tf, fmaxf(c0, c1));
            tn = fmaxf(tn, 0.0f);

            if (tf > tn) {
                const float sigma = expf(sDens[v]);           // from LDS
                const float alpha = 1.0f - expf(-sigma * (tf - tn));
                const float w = T * alpha;
                accR += w * sCol[v * 3 + 0];                  // from LDS
                accG += w * sCol[v * 3 + 1];
                accB += w * sCol[v * 3 + 2];
                accD += w * 0.5f * (tn + tf);
                T *= (1.0f - alpha + 1e-10f);
                if (cnt < MAX_HITS) myidx[cnt] = v;
                ++cnt;
            }

            // Step to next voxel along the smallest boundary-crossing t.
            if (tMx <= tMy && tMx <= tMz) {
                cx += sx; if ((unsigned)cx >= (unsigned)BASE_RES) break; tMx += tDx;
            } else if (tMy <= tMz) {
                cy += sy; if ((unsigned)cy >= (unsigned)BASE_RES) break; tMy += tDy;
            } else {
                cz += sz; if ((unsigned)cz >= (unsigned)BASE_RES) break; tMz += tDz;
            }
        }
    }

    out_rgb[r * 3 + 0] = accR;
    out_rgb[r * 3 + 1] = accG;
    out_rgb[r * 3 + 2] = accB;
    out_depth[r] = accD;
    out_counts[r] = cnt;
    for (int j = cnt; j < MAX_HITS; ++j) myidx[j] = 0;   // d_out is poisoned: pad.
}

extern "C" void kernel_launch(void* const* d_in, const int* in_sizes, int n_in,
                              void* d_out, int out_size, void* d_ws, size_t ws_size,
                              hipStream_t stream) {
    (void)n_in; (void)out_size; (void)d_ws; (void)ws_size;
    const float* ray_o = (const float*)d_in[0];
    const float* ray_d = (const float*)d_in[1];
    const float* vpos  = (const float*)d_in[2];
    const float* vsize = (const float*)d_in[3];
    const float* vdens = (const float*)d_in[4];
    const float* vcol  = (const float*)d_in[5];

    const int R = in_sizes[0] / 3;

    float* out     = (float*)d_out;
    float* o_rgb   = out;                       // R*3
    float* o_depth = out + (size_t)R * 3;       // R
    int*   o_cnt   = (int*)(out + (size_t)R * 4);
    int*   o_idx   = (int*)(out + (size_t)R * 5);

    const int block = 256;
    const int grid  = (R + block - 1) / block;
    svraster_dda_kernel<<<grid, block, 0, stream>>>(
        ray_o, ray_d, vpos, vsize, vdens, vcol,
        o_rgb, o_depth, o_cnt, o_idx, R);
}